// LameGAT_73504070303820
// MI455X (gfx1250) — compile-verified
//
#include <hip/hip_runtime.h>
#include <math.h>

typedef float v2f __attribute__((ext_vector_type(2)));
typedef float v8f __attribute__((ext_vector_type(8)));

#define NEG_SLOPE 0.2f

__device__ __forceinline__ float lrelu(float x) { return x > 0.f ? x : NEG_SLOPE * x; }
__device__ __forceinline__ float eluf(float x)  { return x > 0.f ? x : __expf(x) - 1.f; }

// float atomic max via monotonic int/uint trick (init value is -inf)
__device__ __forceinline__ void atomicMaxF(float* addr, float v) {
  if (v >= 0.f) atomicMax((int*)addr, __float_as_int(v));
  else          atomicMin((unsigned int*)addr, (unsigned int)__float_as_int(v));
}

// ---------------- fill ----------------
__global__ void fill_kernel(float* __restrict__ p, float v, int n) {
  int i = blockIdx.x * blockDim.x + threadIdx.x;
  if (i < n) p[i] = v;
}

// ---------------- GEMM: Y[M,NO] = X[M,K] @ W[NO,K]^T  (fp32 WMMA 16x16x4) ----------------
// One wave per 16(M)x32(N) output tile, two accumulators sharing one A fragment.
// A frag: lane holds A[m0+lane%16][k0+2*(lane/16)+v], v=0,1 (contiguous float2).
// B frag (B[k][n] = W[n][k]): lane holds W[n+lane%16][k0+2*(lane/16)+v].
// Out-of-range rows/cols: pointers clamped to row 0 (loads valid, values ignored),
// stores guarded — no per-iteration EXEC predication in the hot loop.
// C/D frag: acc[r] = Y[m0 + 8*(lane/16) + r][n + lane%16].
__global__ void gemm_xwT_wmma(const float* __restrict__ X, const float* __restrict__ W,
                              float* __restrict__ Y, int M, int K, int NO) {
  const int lane = threadIdx.x & 31;
  const int wave = (blockIdx.x * blockDim.x + threadIdx.x) >> 5;
  const int mtiles = (M + 15) >> 4;
  const int ntiles = (NO + 31) >> 5;          // 32-wide N tiles
  if (wave >= mtiles * ntiles) return;
  const int mt = wave / ntiles, nt = wave - mt * ntiles;
  const int m = mt * 16 + (lane & 15);
  const int mload = (m < M) ? m : 0;          // clamp, store-guarded later
  const int n0 = nt * 32 + (lane & 15);
  const int n1 = n0 + 16;
  const int khalf = (lane >> 4) * 2;
  const float* __restrict__ xrow  = X + (size_t)mload * K + khalf;
  const float* __restrict__ wrow0 = W + (size_t)(n0 < NO ? n0 : 0) * K + khalf;
  const float* __restrict__ wrow1 = W + (size_t)(n1 < NO ? n1 : 0) * K + khalf;
  v8f acc0 = {}, acc1 = {};
#pragma unroll 4
  for (int k0 = 0; k0 < K; k0 += 4) {
    v2f a  = *(const v2f*)(xrow + k0);
    v2f b0 = *(const v2f*)(wrow0 + k0);
    v2f b1 = *(const v2f*)(wrow1 + k0);
    acc0 = __builtin_amdgcn_wmma_f32_16x16x4_f32(false, a, false, b0, (short)0, acc0,
                                                 false, false);
    acc1 = __builtin_amdgcn_wmma_f32_16x16x4_f32(false, a, false, b1, (short)0, acc1,
                                                 false, false);
  }
  const int mrow0 = mt * 16 + (lane >> 4) * 8;
#pragma unroll
  for (int r = 0; r < 8; ++r) {
    int mm = mrow0 + r;
    if (mm < M) {
      if (n0 < NO) Y[(size_t)mm * NO + n0] = acc0[r];
      if (n1 < NO) Y[(size_t)mm * NO + n1] = acc1[r];
    }
  }
}

// ---------------- layer-1 edge scores: H=8, C=32 (256 feats) ----------------
// wave per edge; lane handles 8 contiguous feats; head = lane/4; reduce over 4 lanes.
__global__ void edge_scores_h8(const float* __restrict__ XL, const float* __restrict__ XR,
                               const float* __restrict__ att, const int* __restrict__ src,
                               const int* __restrict__ dst, float* __restrict__ Ebuf,
                               float* __restrict__ emax, int E) {
  const int lane = threadIdx.x & 31;
  const int e = (blockIdx.x * blockDim.x + threadIdx.x) >> 5;
  if (e >= E) return;
  const int s = src[e], d = dst[e];
  const float4* pa = (const float4*)(XL + (size_t)s * 256 + lane * 8);
  const float4* pr = (const float4*)(XR + (size_t)d * 256 + lane * 8);
  const float4* pt = (const float4*)(att + lane * 8);
  float4 a0 = pa[0], a1 = pa[1];
  float4 r0 = pr[0], r1 = pr[1];
  float4 t0 = pt[0], t1 = pt[1];
  float p = t0.x * lrelu(a0.x + r0.x) + t0.y * lrelu(a0.y + r0.y)
          + t0.z * lrelu(a0.z + r0.z) + t0.w * lrelu(a0.w + r0.w)
          + t1.x * lrelu(a1.x + r1.x) + t1.y * lrelu(a1.y + r1.y)
          + t1.z * lrelu(a1.z + r1.z) + t1.w * lrelu(a1.w + r1.w);
  p += __shfl_xor(p, 1, 32);
  p += __shfl_xor(p, 2, 32);
  if ((lane & 3) == 0) {
    int h = lane >> 2;
    Ebuf[(size_t)e * 8 + h] = p;
    atomicMaxF(&emax[(size_t)d * 8 + h], p);
  }
}

// ---------------- layer-2 edge scores: H=1, C=40 ----------------
__global__ void edge_scores_h1c40(const float* __restrict__ XL, const float* __restrict__ XR,
                                  const float* __restrict__ att, const int* __restrict__ src,
                                  const int* __restrict__ dst, float* __restrict__ Ebuf,
                                  float* __restrict__ emax, int E) {
  const int lane = threadIdx.x & 31;
  const int e = (blockIdx.x * blockDim.x + threadIdx.x) >> 5;
  if (e >= E) return;
  const int s = src[e], d = dst[e];
  float p = att[lane] * lrelu(XL[(size_t)s * 40 + lane] + XR[(size_t)d * 40 + lane]);
  if (lane < 8)
    p += att[32 + lane] * lrelu(XL[(size_t)s * 40 + 32 + lane] + XR[(size_t)d * 40 + 32 + lane]);
#pragma unroll
  for (int k = 1; k < 32; k <<= 1) p += __shfl_xor(p, k, 32);
  if (lane == 0) {
    Ebuf[e] = p;
    atomicMaxF(&emax[d], p);
  }
}

// ---------------- p = exp(e - emax[dst]); segment-sum into den ----------------
__global__ void edge_expsum(float* __restrict__ Ebuf, const float* __restrict__ emax,
                            float* __restrict__ den, const int* __restrict__ dst,
                            int E, int H) {
  int i = blockIdx.x * blockDim.x + threadIdx.x;
  if (i >= E * H) return;
  int e = i / H, h = i - e * H;
  int d = dst[e];
  float p = __expf(Ebuf[i] - emax[d * H + h]);
  Ebuf[i] = p;
  atomicAdd(&den[d * H + h], p);
}

// ---------------- layer-1 aggregation: out[dst] += alpha * XL[src] ----------------
__global__ void edge_agg_h8(const float* __restrict__ XL, const float* __restrict__ P,
                            const float* __restrict__ den, const int* __restrict__ src,
                            const int* __restrict__ dst, float* __restrict__ out, int E) {
  const int lane = threadIdx.x & 31;
  const int e = (blockIdx.x * blockDim.x + threadIdx.x) >> 5;
  if (e >= E) return;
  const int s = src[e], d = dst[e];
  const int h = lane >> 2;
  const float alpha = P[(size_t)e * 8 + h] / den[(size_t)d * 8 + h];
  const float4* pa = (const float4*)(XL + (size_t)s * 256 + lane * 8);
  float4 a0 = pa[0], a1 = pa[1];
  float* ob = out + (size_t)d * 256 + lane * 8;
  atomicAdd(ob + 0, alpha * a0.x); atomicAdd(ob + 1, alpha * a0.y);
  atomicAdd(ob + 2, alpha * a0.z); atomicAdd(ob + 3, alpha * a0.w);
  atomicAdd(ob + 4, alpha * a1.x); atomicAdd(ob + 5, alpha * a1.y);
  atomicAdd(ob + 6, alpha * a1.z); atomicAdd(ob + 7, alpha * a1.w);
}

// ---------------- layer-2 aggregation ----------------
__global__ void edge_agg_h1c40(const float* __restrict__ XL, const float* __restrict__ P,
                               const float* __restrict__ den, const int* __restrict__ src,
                               const int* __restrict__ dst, float* __restrict__ out, int E) {
  const int lane = threadIdx.x & 31;
  const int e = (blockIdx.x * blockDim.x + threadIdx.x) >> 5;
  if (e >= E) return;
  const int s = src[e], d = dst[e];
  const float alpha = P[e] / den[d];
  atomicAdd(&out[(size_t)d * 40 + lane], alpha * XL[(size_t)s * 40 + lane]);
  if (lane < 8)
    atomicAdd(&out[(size_t)d * 40 + 32 + lane], alpha * XL[(size_t)s * 40 + 32 + lane]);
}

// ---------------- h = elu(h + b) ----------------
__global__ void bias_elu(float* __restrict__ h, const float* __restrict__ b, int total, int F) {
  int i = blockIdx.x * blockDim.x + threadIdx.x;
  if (i >= total) return;
  float v = h[i] + b[i % F];
  h[i] = eluf(v);
}

// ---------------- final: o = elu(o + b2); out2 = log_softmax(o) ---------------
// wave per node; lane l covers c=l (l<32) and c=32+l (l<8).
__global__ void final_elu_lsm(float* __restrict__ out, const float* __restrict__ b, int N) {
  const int lane = threadIdx.x & 31;
  const int node = (blockIdx.x * blockDim.x + threadIdx.x) >> 5;
  if (node >= N) return;
  float* o = out + (size_t)node * 40;
  float v0 = eluf(o[lane] + b[lane]);
  float v1 = (lane < 8) ? eluf(o[32 + lane] + b[32 + lane]) : -INFINITY;
  float m = fmaxf(v0, v1);
#pragma unroll
  for (int k = 1; k < 32; k <<= 1) m = fmaxf(m, __shfl_xor(m, k, 32));
  float s = __expf(v0 - m) + ((lane < 8) ? __expf(v1 - m) : 0.f);
#pragma unroll
  for (int k = 1; k < 32; k <<= 1) s += __shfl_xor(s, k, 32);
  const float lse = m + logf(s);
  float* l = out + (size_t)N * 40 + (size_t)node * 40;
  o[lane] = v0;
  l[lane] = v0 - lse;
  if (lane < 8) { o[32 + lane] = v1; l[32 + lane] = v1 - lse; }
}

static inline unsigned nblk(long long t, int b) { return (unsigned)((t + b - 1) / b); }

extern "C" void kernel_launch(void* const* d_in, const int* in_sizes, int n_in,
                              void* d_out, int out_size, void* d_ws, size_t ws_size,
                              hipStream_t stream) {
  const float* x    = (const float*)d_in[0];
  const int*   src  = (const int*)d_in[1];
  const int*   dst  = (const int*)d_in[2];
  const float* Wl1  = (const float*)d_in[3];
  const float* Wr1  = (const float*)d_in[4];
  const float* att1 = (const float*)d_in[5];
  const float* b1   = (const float*)d_in[6];
  const float* Wl2  = (const float*)d_in[7];
  const float* Wr2  = (const float*)d_in[8];
  const float* att2 = (const float*)d_in[9];
  const float* b2   = (const float*)d_in[10];
  const int N = in_sizes[0] / 128;   // 50000
  const int E = in_sizes[1];         // 650000
  float* out = (float*)d_out;

  // workspace layout (floats); H1 aliases XR1 (XR1 dead after layer-1 scores)
  float* ws    = (float*)d_ws;
  float* XL1   = ws;                          // N*256
  float* XR1   = XL1 + (size_t)N * 256;       // N*256
  float* H1    = XR1;                         // alias
  float* E1    = XR1 + (size_t)N * 256;       // E*8
  float* emax1 = E1 + (size_t)E * 8;          // N*8
  float* den1  = emax1 + (size_t)N * 8;       // N*8
  float* XL2   = den1 + (size_t)N * 8;        // N*40
  float* XR2   = XL2 + (size_t)N * 40;        // N*40
  float* E2    = XR2 + (size_t)N * 40;        // E
  float* emax2 = E2 + (size_t)E;              // N
  float* den2  = emax2 + (size_t)N;           // N

  const int B = 256;

  // ---- layer 1: dense transforms via WMMA (16x32 tile per wave) ----
  {
    long long waves = (long long)((N + 15) / 16) * ((256 + 31) / 32);
    gemm_xwT_wmma<<<nblk(waves * 32, B), B, 0, stream>>>(x, Wl1, XL1, N, 128, 256);
    gemm_xwT_wmma<<<nblk(waves * 32, B), B, 0, stream>>>(x, Wr1, XR1, N, 128, 256);
  }
  fill_kernel<<<nblk((long long)N * 8, B), B, 0, stream>>>(emax1, -INFINITY, N * 8);
  fill_kernel<<<nblk((long long)N * 8, B), B, 0, stream>>>(den1, 0.f, N * 8);
  edge_scores_h8<<<nblk((long long)E * 32, B), B, 0, stream>>>(XL1, XR1, att1, src, dst,
                                                               E1, emax1, E);
  edge_expsum<<<nblk((long long)E * 8, B), B, 0, stream>>>(E1, emax1, den1, dst, E, 8);
  fill_kernel<<<nblk((long long)N * 256, B), B, 0, stream>>>(H1, 0.f, N * 256);
  edge_agg_h8<<<nblk((long long)E * 32, B), B, 0, stream>>>(XL1, E1, den1, src, dst, H1, E);
  bias_elu<<<nblk((long long)N * 256, B), B, 0, stream>>>(H1, b1, N * 256, 256);

  // ---- layer 2: dense transforms via WMMA (NO=40, clamped/guarded tiles) ----
  {
    long long waves = (long long)((N + 15) / 16) * ((40 + 31) / 32);
    gemm_xwT_wmma<<<nblk(waves * 32, B), B, 0, stream>>>(H1, Wl2, XL2, N, 256, 40);
    gemm_xwT_wmma<<<nblk(waves * 32, B), B, 0, stream>>>(H1, Wr2, XR2, N, 256, 40);
  }
  fill_kernel<<<nblk((long long)N, B), B, 0, stream>>>(emax2, -INFINITY, N);
  fill_kernel<<<nblk((long long)N, B), B, 0, stream>>>(den2, 0.f, N);
  edge_scores_h1c40<<<nblk((long long)E * 32, B), B, 0, stream>>>(XL2, XR2, att2, src, dst,
                                                                  E2, emax2, E);
  edge_expsum<<<nblk((long long)E, B), B, 0, stream>>>(E2, emax2, den2, dst, E, 1);
  fill_kernel<<<nblk((long long)N * 40, B), B, 0, stream>>>(out, 0.f, N * 40);
  edge_agg_h1c40<<<nblk((long long)E * 32, B), B, 0, stream>>>(XL2, E2, den2, src, dst, out, E);
  final_elu_lsm<<<nblk((long long)N * 32, B), B, 0, stream>>>(out, b2, N);
}